// SimilarityFewShotSegmenter_14920716386602
// MI455X (gfx1250) — compile-verified
//
#include <hip/hip_runtime.h>
#include <hip/hip_bf16.h>

typedef float v2f __attribute__((ext_vector_type(2)));
typedef float v8f __attribute__((ext_vector_type(8)));

#define B_     2
#define M_     9
#define MS_    8      // support count
#define D_     256
#define HW_    2304   // 48*48
#define NT_    144    // HW_/16 pixel tiles
#define KS_    64     // D_/4 K-steps per WMMA chain
#define QB_    18     // q-blocks: NT_/8 waves
#define SPLITS_ 16
#define TPS_   ((MS_ * NT_) / SPLITS_)   // 72 support tiles per split
#define OUTHW_ 224
#define TILE_F 4096                       // floats per 16-pixel fragment tile
#define LPSTRIDE_ (B_ * 4 * HW_)

// ---------------------------------------------------------------------------
// Pass 1: normalize along D, emit WMMA fragment order:
// frag[((bm*NT_+tile)*KS_ + ks)*64 + lane*2 + j]
//   = emb_norm[d = ks*4 + (lane>>4)*2 + j, p = tile*16 + (lane&15)]
// ---------------------------------------------------------------------------
__global__ void prep_norm_frag(const float* __restrict__ emb,
                               float* __restrict__ frag) {
  const int blk  = blockIdx.x;         // (b*9+m)*NT_ + tile
  const int tile = blk % NT_;
  const int bm   = blk / NT_;
  const float* e = emb + (size_t)bm * D_ * HW_ + tile * 16;
  float* out     = frag + (size_t)blk * TILE_F;

  __shared__ float raw[D_ * 16];       // 16 KB
  __shared__ float part[256];
  __shared__ float rinv[16];

  const int t = threadIdx.x;
#pragma unroll
  for (int i = 0; i < 16; ++i) {
    const int d  = i * 16 + (t >> 4);
    const int pl = t & 15;
    raw[d * 16 + pl] = e[(size_t)d * HW_ + pl];
  }
  __syncthreads();
  {
    const int pl = t & 15, dc = t >> 4;
    float s = 0.f;
#pragma unroll
    for (int k = 0; k < 16; ++k) {
      const float v = raw[(dc * 16 + k) * 16 + pl];
      s += v * v;
    }
    part[t] = s;
  }
  __syncthreads();
  if (t < 16) {
    float s = 0.f;
#pragma unroll
    for (int k = 0; k < 16; ++k) s += part[k * 16 + t];
    rinv[t] = 1.0f / fmaxf(sqrtf(s), 1e-12f);
  }
  __syncthreads();
#pragma unroll
  for (int i = 0; i < 16; ++i) {
    const int f    = i * 256 + t;
    const int j    = f & 1;
    const int lane = (f >> 1) & 31;
    const int ks   = f >> 6;
    const int d    = ks * 4 + (lane >> 4) * 2 + j;
    const int pl   = lane & 15;
    out[f] = raw[d * 16 + pl] * rinv[pl];
  }
}

// ---------------------------------------------------------------------------
// Pass 2: pack 4 class-mask bits per (b, support, pixel); bit0 = background.
// ---------------------------------------------------------------------------
__global__ void prep_mask(const float* __restrict__ pm,
                          unsigned* __restrict__ mbits) {
  const int idx = blockIdx.x * blockDim.x + threadIdx.x;
  if (idx >= B_ * MS_ * HW_) return;
  const int p  = idx % HW_;
  const int bm = idx / HW_;
  const float* base = pm + (size_t)bm * 4 * HW_ + p;
  const float v1 = base[1 * HW_], v2 = base[2 * HW_], v3 = base[3 * HW_];
  unsigned bits = 0u;
  if (v1 + v2 + v3 == 0.f) bits |= 1u;
  if (v1 != 0.f) bits |= 2u;
  if (v2 != 0.f) bits |= 4u;
  if (v3 != 0.f) bits |= 8u;
  mbits[idx] = bits;
}

// ---------------------------------------------------------------------------
// Pass 3: fused GEMM + masked max.
//  Block = (b, q-block of 128 queries, support split), 8 wave32 waves.
//  A fragments (16 queries x 256 K) live in 128 VGPRs per wave.
//  B tiles (16 KB) are double-buffered in LDS via global_load_async_to_lds,
//  shared by all 8 waves; prefetch of tile t+1 overlaps compute of tile t.
// ---------------------------------------------------------------------------
__device__ __forceinline__ void issue_async_tile(unsigned ldsBase,
                                                 const float* gbase, int t) {
#pragma unroll
  for (int i = 0; i < 4; ++i) {
    const unsigned voff = (unsigned)(i * 256 + t) * 16u;  // 16B chunks
    asm volatile("global_load_async_to_lds_b128 %0, %1, %2"
                 :
                 : "v"(ldsBase + voff), "v"(voff), "s"(gbase)
                 : "memory");
  }
}

__global__ void __launch_bounds__(256, 1)
sim_max_kernel(const float* __restrict__ frag,
               const unsigned* __restrict__ mbits,
               float* __restrict__ lpart) {
  const int blk = blockIdx.x;                  // ((b*QB_ + qb)*SPLITS_ + sp)
  const int sp  = blk % SPLITS_;
  const int qb  = (blk / SPLITS_) % QB_;
  const int b   = blk / (SPLITS_ * QB_);
  const int t    = threadIdx.x;
  const int w    = t >> 5;
  const int lane = t & 31;
  const int col  = lane & 15;

  __shared__ float bbuf[2][TILE_F];            // 2 x 16 KB double buffer

  // ---- preload this wave's A tile (16 queries x 256 K) into registers ----
  const int qtile = qb * 8 + w;                // 0..143
  const float* aBase = frag + ((size_t)(b * M_) * NT_ + qtile) * TILE_F;
  v2f aReg[KS_];
#pragma unroll
  for (int ks = 0; ks < KS_; ++ks)
    aReg[ks] = *(const v2f*)(aBase + ks * 64 + lane * 2);

  float runmax[8][4];
#pragma unroll
  for (int i = 0; i < 8; ++i)
#pragma unroll
    for (int n = 0; n < 4; ++n) runmax[i][n] = -__builtin_inff();

  // support tile list for this split
  const int st0 = sp * TPS_;
  const float* sFragBase =
      frag + ((size_t)(b * M_ + 1) * NT_) * TILE_F;   // base of support frags

  // prologue: async-prefetch tile 0 into buffer 0
  {
    const unsigned lds0 = (unsigned)(uintptr_t)&bbuf[0][0];
    issue_async_tile(lds0, sFragBase + (size_t)st0 * TILE_F, t);
  }

  for (int ti = 0; ti < TPS_; ++ti) {
    const int st = st0 + ti;                   // global support tile 0..1151
    const int ms = st / NT_;
    const int kt = st - ms * NT_;

    // wait for this tile's async loads (issued by this wave), then block
    // barrier so every wave's portion is visible and the other buffer is free
    asm volatile("s_wait_asynccnt 0x0" ::: "memory");
    __syncthreads();

    if (ti + 1 < TPS_) {
      const unsigned ldsN = (unsigned)(uintptr_t)&bbuf[(ti + 1) & 1][0];
      issue_async_tile(ldsN, sFragBase + (size_t)(st + 1) * TILE_F, t);
    }

    const unsigned bits = mbits[(size_t)(b * MS_ + ms) * HW_ + kt * 16 + col];
    const float* bptr = &bbuf[ti & 1][0];

    v8f acc = {0.f, 0.f, 0.f, 0.f, 0.f, 0.f, 0.f, 0.f};
#pragma unroll
    for (int ks = 0; ks < KS_; ++ks) {
      const v2f bv = *(const v2f*)(bptr + ks * 64 + lane * 2);  // ds_load_b64
      acc = __builtin_amdgcn_wmma_f32_16x16x4_f32(
          false, aReg[ks], false, bv, (short)0, acc, false, false);
    }
    // C/D layout: VGPR i -> row i (lanes 0-15) / row i+8 (lanes 16-31), N=col
#pragma unroll
    for (int i = 0; i < 8; ++i) {
      const float sim = acc[i];
#pragma unroll
      for (int n = 0; n < 4; ++n)
        if ((bits >> n) & 1u) runmax[i][n] = fmaxf(runmax[i][n], sim);
    }
  }

  // reduce over the 16 column slots inside each half-wave
#pragma unroll
  for (int i = 0; i < 8; ++i)
#pragma unroll
    for (int n = 0; n < 4; ++n) {
      float v = runmax[i][n];
      for (int off = 8; off >= 1; off >>= 1)
        v = fmaxf(v, __shfl_xor(v, off, 16));
      runmax[i][n] = v;
    }

  // waves own disjoint queries: lanes 0/16 write 8 rows x 4 classes each
  if (col == 0) {
    const int rbase = (lane >> 4) * 8;
    float* dst = lpart + (size_t)sp * LPSTRIDE_ + (size_t)(b * 4) * HW_ +
                 qtile * 16;
#pragma unroll
    for (int i = 0; i < 8; ++i)
#pragma unroll
      for (int n = 0; n < 4; ++n)
        dst[(size_t)n * HW_ + rbase + i] = runmax[i][n];
  }
}

// ---------------------------------------------------------------------------
// Pass 4: max-combine the 16 support splits + half-pixel bilinear 48 -> 224.
// ---------------------------------------------------------------------------
__device__ __forceinline__ float split_max(const float* L, int off) {
  float v = L[off];
#pragma unroll
  for (int s = 1; s < SPLITS_; ++s)
    v = fmaxf(v, L[(size_t)s * LPSTRIDE_ + off]);
  return v;
}

__global__ void resize_kernel(const float* __restrict__ lpart,
                              float* __restrict__ out) {
  const int idx = blockIdx.x * blockDim.x + threadIdx.x;
  if (idx >= B_ * 4 * OUTHW_ * OUTHW_) return;
  const int x  = idx % OUTHW_;
  const int y  = (idx / OUTHW_) % OUTHW_;
  const int bn = idx / (OUTHW_ * OUTHW_);
  const int n  = bn & 3;

  const float scale = 48.0f / (float)OUTHW_;
  const float sx = ((float)x + 0.5f) * scale - 0.5f;
  const float sy = ((float)y + 0.5f) * scale - 0.5f;
  int x0 = (int)floorf(sx), y0 = (int)floorf(sy);
  const float fx = sx - (float)x0, fy = sy - (float)y0;
  int x1 = min(max(x0 + 1, 0), 47), y1 = min(max(y0 + 1, 0), 47);
  x0 = min(max(x0, 0), 47);
  y0 = min(max(y0, 0), 47);

  const int base = bn * HW_;
  const float w00 = (1.f - fx) * (1.f - fy), w10 = fx * (1.f - fy);
  const float w01 = (1.f - fx) * fy,         w11 = fx * fy;
  float acc = 0.f;
  if (w00 > 0.f) acc += w00 * split_max(lpart, base + y0 * 48 + x0);
  if (w10 > 0.f) acc += w10 * split_max(lpart, base + y0 * 48 + x1);
  if (w01 > 0.f) acc += w01 * split_max(lpart, base + y1 * 48 + x0);
  if (w11 > 0.f) acc += w11 * split_max(lpart, base + y1 * 48 + x1);
  if (n == 0 && acc == -__builtin_inff()) acc = 0.f;
  out[idx] = acc;
}

// ---------------------------------------------------------------------------
extern "C" void kernel_launch(void* const* d_in, const int* in_sizes, int n_in,
                              void* d_out, int out_size, void* d_ws,
                              size_t ws_size, hipStream_t stream) {
  const float* emb = (const float*)d_in[0];   // (2,9,256,48,48) f32
  const float* pm  = (const float*)d_in[1];   // (2,8,4,48,48)  f32
  float* out       = (float*)d_out;           // (2,4,224,224)  f32

  char* ws = (char*)d_ws;
  const size_t fragElems = (size_t)B_ * M_ * NT_ * TILE_F;  // 10,616,832
  float*    frag  = (float*)ws;
  unsigned* mbits = (unsigned*)(ws + fragElems * sizeof(float));
  float*    lpart = (float*)(ws + fragElems * sizeof(float) +
                             (size_t)B_ * MS_ * HW_ * sizeof(unsigned));

  prep_norm_frag<<<B_ * M_ * NT_, 256, 0, stream>>>(emb, frag);
  prep_mask<<<(B_ * MS_ * HW_ + 255) / 256, 256, 0, stream>>>(pm, mbits);
  sim_max_kernel<<<B_ * QB_ * SPLITS_, 256, 0, stream>>>(frag, mbits, lpart);
  resize_kernel<<<(B_ * 4 * OUTHW_ * OUTHW_ + 255) / 256, 256, 0, stream>>>(
      lpart, out);
}